// LossModule_60266981097717
// MI455X (gfx1250) — compile-verified
//
#include <hip/hip_runtime.h>

typedef __attribute__((ext_vector_type(2))) float v2f;
typedef __attribute__((ext_vector_type(8))) float v8f;

#define B_ 8
#define N_ 256
#define M_ 16
#define D_ 256

__global__ void zero_out_kernel(float* out) { out[0] = 0.0f; }

// C[b,k,d] = mean_i E[b,k,i,d]; one thread per (b,k,d)
__global__ __launch_bounds__(256) void centroid_kernel(const float* __restrict__ E,
                                                       float* __restrict__ C) {
    int idx = blockIdx.x * blockDim.x + threadIdx.x;   // [0, B*N*D)
    if (idx >= B_ * N_ * D_) return;
    int d  = idx & (D_ - 1);
    int bk = idx >> 8;                                  // idx / D_
    const float* p = E + (size_t)bk * M_ * D_ + d;
    float s = 0.0f;
#pragma unroll
    for (int i = 0; i < M_; ++i) s += p[i * D_];
    C[idx] = s * (1.0f / M_);
}

// One wave per (b, j): rows = e[b,j,0..15,:], columns k = 0..255.
// A fragments resident in VGPRs for the whole wave; B double-buffered in
// groups of 8 K-steps so global_load latency overlaps the WMMA stream.
__global__ __launch_bounds__(256) void loss_kernel(const float* __restrict__ E,
                                                   const float* __restrict__ C,
                                                   const float* __restrict__ wp,
                                                   const float* __restrict__ bp,
                                                   float* __restrict__ out) {
    const float w    = wp[0];
    const float bias = bp[0];

    const int lane = threadIdx.x & 31;
    const int wid  = blockIdx.x * (blockDim.x >> 5) + (threadIdx.x >> 5); // [0, B_*N_)
    const int b    = wid >> 8;
    const int j    = wid & (N_ - 1);
    const int half = lane >> 4;   // K-half for A/B fragments
    const int n    = lane & 15;   // row m for A, column n for B/C/D

    // ---- Load the whole A tile (16x256) into registers: 64 K-step fragments
    const float* Arow = E + (((size_t)(b * N_ + j)) * M_ + n) * D_ + half * 2;
    v2f af[64];
#pragma unroll
    for (int s = 0; s < 64; ++s) af[s] = *(const v2f*)(Arow + 4 * s);

    // ---- Squared norms from the resident A fragments (no extra global reads).
    // Lane (m + 16*h) holds row m's elements at d = 4s + 2h + {0,1}: half the row.
    float psq = 0.0f;
#pragma unroll
    for (int s = 0; s < 64; ++s) psq += af[s].x * af[s].x + af[s].y * af[s].y;
    float sqn[8];
#pragma unroll
    for (int r = 0; r < 8; ++r) {
        int m = r + 8 * half;                     // row this lane's acc[r] belongs to
        sqn[r] = __shfl(psq, m, 32) + __shfl(psq, m + 16, 32);
    }

    float mx[8], sm[8];
#pragma unroll
    for (int r = 0; r < 8; ++r) { mx[r] = -1e30f; sm[r] = 0.0f; }
    float selfacc = 0.0f;

    const int jt = j >> 4;     // column tile containing the diagonal
    const int jn = j & 15;     // lane column within that tile

    const float* Bcol = C + ((size_t)(b * N_ + n)) * D_ + half * 2;  // k = kt*16 + n

    for (int kt = 0; kt < 16; ++kt) {
        const float* Brow = Bcol + (size_t)kt * 16 * D_;
        v2f bf[2][8];
#pragma unroll
        for (int u = 0; u < 8; ++u) bf[0][u] = *(const v2f*)(Brow + 4 * u);

        v8f acc = {};
#pragma unroll
        for (int g = 0; g < 8; ++g) {             // 8 groups x 8 K-steps = D/4
            if (g < 7) {
#pragma unroll
                for (int u = 0; u < 8; ++u)
                    bf[(g + 1) & 1][u] = *(const v2f*)(Brow + 32 * (g + 1) + 4 * u);
            }
#pragma unroll
            for (int u = 0; u < 8; ++u)
                acc = __builtin_amdgcn_wmma_f32_16x16x4_f32(
                    false, af[8 * g + u], false, bf[g & 1][u], (short)0, acc,
                    false, false);
        }

        const bool selfcol = (kt == jt) && (n == jn);
#pragma unroll
        for (int r = 0; r < 8; ++r) {
            float dot = acc[r];                   // row m = r + 8*half, col kt*16 + n
            float v   = w * dot + bias;
            if (selfcol) {                        // leave-one-out centroid on diagonal
                v = w * (M_ * dot - sqn[r]) * (1.0f / (M_ - 1)) + bias;
                selfacc += v;
            }
            float nm = fmaxf(mx[r], v);
            sm[r] = sm[r] * __expf(mx[r] - nm) + __expf(v - nm);
            mx[r] = nm;
        }
    }

    // Merge logsumexp state across the 16 lanes of each half (the n dimension)
#pragma unroll
    for (int off = 1; off < 16; off <<= 1) {
#pragma unroll
        for (int r = 0; r < 8; ++r) {
            float omx = __shfl_xor(mx[r], off, 32);
            float osm = __shfl_xor(sm[r], off, 32);
            float nm  = fmaxf(mx[r], omx);
            sm[r] = sm[r] * __expf(mx[r] - nm) + osm * __expf(omx - nm);
            mx[r] = nm;
        }
    }

    float t = -selfacc;                  // subtract Σ S_self (only diag lanes nonzero)
    if (n == 0) {                        // one lane per half contributes the 8 row LSEs
#pragma unroll
        for (int r = 0; r < 8; ++r) t += mx[r] + __logf(sm[r]);
    }
    // Full-wave sum, then one atomic per wave
#pragma unroll
    for (int off = 1; off < 32; off <<= 1) t += __shfl_xor(t, off, 32);
    if (lane == 0) atomicAdd(out, t);
}

extern "C" void kernel_launch(void* const* d_in, const int* in_sizes, int n_in,
                              void* d_out, int out_size, void* d_ws, size_t ws_size,
                              hipStream_t stream) {
    const float* E  = (const float*)d_in[0];   // [8,256,16,256] fp32
    const float* wp = (const float*)d_in[1];   // scalar
    const float* bp = (const float*)d_in[2];   // scalar
    float* out = (float*)d_out;                // scalar loss
    float* Cws = (float*)d_ws;                 // centroids: B*N*D floats = 2 MB

    zero_out_kernel<<<1, 1, 0, stream>>>(out);

    int ctot = B_ * N_ * D_;
    centroid_kernel<<<(ctot + 255) / 256, 256, 0, stream>>>(E, Cws);

    // one wave per (b,j): B*N = 2048 waves, 8 waves (256 threads) per block
    loss_kernel<<<(B_ * N_) / 8, 256, 0, stream>>>(E, Cws, wp, bp, out);
}